// PositionalAttention_50362786513235
// MI455X (gfx1250) — compile-verified
//
#include <hip/hip_runtime.h>

// PositionalAttention (B=4, C=64, N=64*64=4096), fp32.
// Flash-attention formulation: Q=K=V=X^T, P = softmax(X^T X), out = X P^T + x.
// - f32 WMMA (V_WMMA_F32_16X16X4_F32) so accumulation matches the fp32 reference.
// - Tensor Data Mover (tensor_load_to_lds) stages K/V tiles into LDS with the
//   bank-conflict-free 16+1 padded row layout, double-buffered on TENSORcnt.

typedef __attribute__((ext_vector_type(2))) float v2f;
typedef __attribute__((ext_vector_type(8))) float v8f;
typedef __attribute__((ext_vector_type(4))) unsigned v4u;
typedef __attribute__((ext_vector_type(8))) int v8i;
typedef __attribute__((ext_vector_type(4))) int v4i;

#define BB 4
#define CC 64
#define NN 4096
#define CN (CC * NN)

__device__ __forceinline__ v8f wmma4(v2f a, v2f b, v8f c) {
  // (neg_a, A, neg_b, B, c_mod, C, reuse_a, reuse_b)
  return __builtin_amdgcn_wmma_f32_16x16x4_f32(false, a, false, b, (short)0, c,
                                               false, false);
}

// Issue one TDM descriptor: load a 64(C) x 16(j) fp32 tile of X[b] into LDS at
// lds_addr, inserting 1 pad DWORD after every 16 DWORDs (row stride 17).
__device__ __forceinline__ void tdm_load_tile(unsigned lds_addr,
                                              const float* gptr) {
  unsigned g_lo = (unsigned)__builtin_amdgcn_readfirstlane(
      (unsigned)(unsigned long long)(uintptr_t)gptr);
  unsigned g_hi = (unsigned)__builtin_amdgcn_readfirstlane(
      (unsigned)(((unsigned long long)(uintptr_t)gptr) >> 32));
  unsigned lds = (unsigned)__builtin_amdgcn_readfirstlane(lds_addr);

  // D# group 0: count=1 | lds_addr | global_addr[31:0] | global_addr[56:32], type=2
  v4u g0 = {1u, lds, g_lo, (g_hi & 0x01FFFFFFu) | (2u << 30)};
  // D# group 1:
  //  d0: data_size=2 (4B) <<16 | pad_enable<<20 | pad_interval=3 (16 DWORDs)<<22
  //      | pad_amount=0 (1 DWORD)<<25
  //  d1: tensor_dim0[15:0]=4096 <<16          (bits 79:48 = tensor_dim0)
  //  d2: tensor_dim0[31:16] | tensor_dim1=64 <<16
  //  d3: tensor_dim1[31:16] | tile_dim0=16 <<16
  //  d4: tile_dim1=64 | tile_dim2=0 <<16
  //  d5: tensor_dim0_stride[31:0]=4096
  //  d6,d7: stride high bits / dim1 stride = 0
  v8i g1 = {(int)((2u << 16) | (1u << 20) | (3u << 22)),
            (int)(4096u << 16),
            (int)((4096u >> 16) | (64u << 16)),
            (int)(16u << 16),
            (int)64u,
            (int)4096u,
            0,
            0};
  v4i g2 = {0, 0, 0, 0};
  v4i g3 = {0, 0, 0, 0};
  v8i g4 = {0, 0, 0, 0, 0, 0, 0, 0};
  // 6-arg toolchain variant: (g0, g1, g2, g3, g4, cpol)
  __builtin_amdgcn_tensor_load_to_lds(g0, g1, g2, g3, g4, 0);
}

__launch_bounds__(128)
__global__ void pos_attn_flash_kernel(const float* __restrict__ x,
                                      float* __restrict__ out) {
  // smem layout (floats):
  //   [0    .. 4351] : K/V tile buffer 0, per wave 64 x 17 (padded)
  //   [4352 .. 8703] : K/V tile buffer 1, per wave 64 x 17
  //   [8704 .. 9791] : per-wave P transpose buffers, 16 x 17 each
  // epilogue reuse:
  //   [0    .. 4095] : per-wave unscaled acc partials 4*16*64
  //   [4096 .. 4159] : per-wave row-max m
  //   [4160 .. 4223] : per-wave row-sum l
  __shared__ float smem[9792];

  const int tid  = threadIdx.x;
  const int lane = tid & 31;
  const int w    = tid >> 5;   // wave id 0..3
  const int half = lane >> 4;  // lane group (0: lanes 0-15, 1: lanes 16-31)
  const int ml   = lane & 15;

  const int bid   = blockIdx.x;
  const int b     = bid >> 8;  // 256 i-tiles per batch
  const int itile = bid & 255;
  const int ibase = itile * 16;

  const float* xb = x + b * CN;
  float* ob       = out + b * CN;

  float* kvbuf0 = smem + w * 1088;
  float* kvbuf1 = smem + 4 * 1088 + w * 1088;
  float* pb     = smem + 8 * 1088 + w * (16 * 17);
  const unsigned kv0_addr = (unsigned)(uintptr_t)kvbuf0;  // LDS byte offset
  const unsigned kv1_addr = (unsigned)(uintptr_t)kvbuf1;

  // ---- Load Q fragments once (A-matrix layout for 16x16x4 f32 WMMA).
  // A is 16x4: lanes 0-15 hold K=(0,1) in (v0,v1); lanes 16-31 hold K=(2,3).
  v2f qa[16];
#pragma unroll
  for (int s = 0; s < 16; ++s) {
    const int c0 = 4 * s + 2 * half;
    qa[s].x = xb[c0 * NN + ibase + ml];
    qa[s].y = xb[(c0 + 1) * NN + ibase + ml];
  }

  // Online-softmax state. Each lane-half owns 8 rows (C/D layout: VGPR r -> row r + 8*half).
  v8f acc[4];
  float mrow[8], lrow[8];
#pragma unroll
  for (int r = 0; r < 8; ++r) {
    mrow[r] = -3.0e38f;
    lrow[r] = 0.0f;
  }
#pragma unroll
  for (int cb = 0; cb < 4; ++cb) acc[cb] = v8f{};

  const int jstart = w * (NN / 4);  // each wave owns a 1024-wide j stripe

  // Prologue: DMA tile 0 into buffer 0.
  tdm_load_tile(kv0_addr, xb + jstart);

  for (int t = 0; t < 64; ++t) {
    float* kv = (t & 1) ? kvbuf1 : kvbuf0;

    if (t < 63) {
      // WAR guard: our own DS reads of the other buffer (iteration t-1) must
      // retire before the TDM engine refills it.
      asm volatile("s_wait_dscnt 0x0" ::: "memory");
      tdm_load_tile((t & 1) ? kv0_addr : kv1_addr,
                    xb + jstart + (t + 1) * 16);
      // TDM completes in order per wave: <=1 outstanding => tile t is resident.
      __builtin_amdgcn_s_wait_tensorcnt((short)1);
    } else {
      __builtin_amdgcn_s_wait_tensorcnt((short)0);
    }
    asm volatile("" ::: "memory");

    // ---- S = Q K^T : 16 k-steps over C=64.
    // B is 4x16: v0 holds K rows (0 | 2) across lane halves, v1 holds (1 | 3).
    v8f s8 = v8f{};
#pragma unroll
    for (int s = 0; s < 16; ++s) {
      const int cb0 = 4 * s + 2 * half;
      v2f bf;
      bf.x = kv[cb0 * 17 + ml];
      bf.y = kv[(cb0 + 1) * 17 + ml];
      s8 = wmma4(qa[s], bf, s8);
    }

    // ---- Online softmax per row; write P into LDS in row-major (m, n) form.
#pragma unroll
    for (int r = 0; r < 8; ++r) {
      float v = s8[r];
      float rmax = v;
      rmax = fmaxf(rmax, __shfl_xor(rmax, 1, 32));
      rmax = fmaxf(rmax, __shfl_xor(rmax, 2, 32));
      rmax = fmaxf(rmax, __shfl_xor(rmax, 4, 32));
      rmax = fmaxf(rmax, __shfl_xor(rmax, 8, 32));
      const float mnew  = fmaxf(mrow[r], rmax);
      const float p     = __expf(v - mnew);
      const float scale = __expf(mrow[r] - mnew);
      float rsum = p;
      rsum += __shfl_xor(rsum, 1, 32);
      rsum += __shfl_xor(rsum, 2, 32);
      rsum += __shfl_xor(rsum, 4, 32);
      rsum += __shfl_xor(rsum, 8, 32);
      lrow[r] = lrow[r] * scale + rsum;
      mrow[r] = mnew;
#pragma unroll
      for (int cb = 0; cb < 4; ++cb) acc[cb][r] *= scale;
      pb[(r + 8 * half) * 17 + ml] = p;  // P[m][n]
    }
    // Intra-wave LDS exchange: per-wave DS ops are processed in order; pin the
    // compiler's ordering across the store->load transpose.
    __builtin_amdgcn_wave_barrier();

    // ---- O += P V : A = P (16x16 as 4 k-steps of 16x4), B = V tile (j x c).
#pragma unroll
    for (int cb = 0; cb < 4; ++cb) {
      const int col = cb * 16 + ml;  // output channel for this lane
#pragma unroll
      for (int kk = 0; kk < 4; ++kk) {
        const int j0 = 4 * kk + 2 * half;
        v2f af, bf;
        af.x = pb[ml * 17 + j0];   // P[m = lane][k = j0]
        af.y = pb[ml * 17 + j0 + 1];
        bf.x = kv[col * 17 + j0];  // V[j = j0][c = col] = X[c][jb + j0]
        bf.y = kv[col * 17 + j0 + 1];
        acc[cb] = wmma4(af, bf, acc[cb]);
      }
    }
  }

  __syncthreads();  // all waves done with kv/pb; reuse smem for the merge

  // ---- Dump unscaled partials + per-row (m, l) stats.
#pragma unroll
  for (int cb = 0; cb < 4; ++cb) {
#pragma unroll
    for (int r = 0; r < 8; ++r) {
      const int i = r + 8 * half;
      const int c = cb * 16 + ml;
      smem[(w * 16 + i) * 64 + c] = acc[cb][r];
    }
  }
  if (ml == 0) {
#pragma unroll
    for (int r = 0; r < 8; ++r) {
      const int i = r + 8 * half;
      smem[4096 + w * 16 + i] = mrow[r];
      smem[4160 + w * 16 + i] = lrow[r];
    }
  }
  __syncthreads();

  // ---- Log-sum-exp merge of 4 wave partials, add residual, store (i fastest).
#pragma unroll
  for (int k = 0; k < 8; ++k) {
    const int e = tid + k * 128;  // e = c*16 + i, 1024 elements
    const int i = e & 15;
    const int c = e >> 4;
    float M = -3.0e38f;
#pragma unroll
    for (int ww = 0; ww < 4; ++ww) M = fmaxf(M, smem[4096 + ww * 16 + i]);
    float L = 0.0f, A = 0.0f;
#pragma unroll
    for (int ww = 0; ww < 4; ++ww) {
      const float sc = __expf(smem[4096 + ww * 16 + i] - M);
      L += smem[4160 + ww * 16 + i] * sc;
      A += smem[(ww * 16 + i) * 64 + c] * sc;
    }
    ob[c * NN + ibase + i] = A / L + xb[c * NN + ibase + i];
  }
}

extern "C" void kernel_launch(void* const* d_in, const int* in_sizes, int n_in,
                              void* d_out, int out_size, void* d_ws, size_t ws_size,
                              hipStream_t stream) {
  (void)in_sizes; (void)n_in; (void)out_size; (void)d_ws; (void)ws_size;
  const float* x = (const float*)d_in[0];
  float* out = (float*)d_out;
  const int blocks = BB * (NN / 16);  // 1024 workgroups, one per (batch, 16-row i-tile)
  pos_attn_flash_kernel<<<blocks, 128, 0, stream>>>(x, out);
}